// KMeans_2723009266535
// MI455X (gfx1250) — compile-verified
//
#include <hip/hip_runtime.h>
#include <cstdint>
#include <cstddef>

#define BATCH 4
#define NPTS  16384
#define DIM   256
#define KC    512
#define NITER 10
#define EPSF  1e-16f
#define ROWS  64          // points per block in assign kernel
#define TPBA  128         // 4 waves of 32
#define PITCH (DIM + 2)   // LDS row pitch (floats), breaks 64-bank conflicts

typedef float v2f __attribute__((ext_vector_type(2)));
typedef float v8f __attribute__((ext_vector_type(8)));

// ---------------------------------------------------------------------------
// MT19937 reproducing CPython random.seed(42); random.sample(range(16384),512)
// ---------------------------------------------------------------------------
__device__ static unsigned mt_next(unsigned* mt, int& mti) {
  if (mti >= 624) {
    for (int kk = 0; kk < 624; ++kk) {
      unsigned y = (mt[kk] & 0x80000000u) | (mt[(kk + 1) % 624] & 0x7fffffffu);
      mt[kk] = mt[(kk + 397) % 624] ^ (y >> 1) ^ ((y & 1u) ? 0x9908b0dfu : 0u);
    }
    mti = 0;
  }
  unsigned y = mt[mti++];
  y ^= y >> 11;
  y ^= (y << 7) & 0x9d2c5680u;
  y ^= (y << 15) & 0xefc60000u;
  y ^= y >> 18;
  return y;
}

__global__ void k_init_inds(int* __restrict__ inds) {
  if (threadIdx.x != 0 || blockIdx.x != 0) return;
  unsigned mt[624];
  mt[0] = 19650218u;                                   // init_genrand(19650218)
  for (int i = 1; i < 624; ++i)
    mt[i] = 1812433253u * (mt[i - 1] ^ (mt[i - 1] >> 30)) + (unsigned)i;
  int i = 1, j = 0;                                    // init_by_array([42])
  for (int k = 624; k; --k) {
    mt[i] = (mt[i] ^ ((mt[i - 1] ^ (mt[i - 1] >> 30)) * 1664525u)) + 42u + (unsigned)j;
    ++i; ++j;
    if (i >= 624) { mt[0] = mt[623]; i = 1; }
    if (j >= 1) j = 0;
  }
  for (int k = 623; k; --k) {
    mt[i] = (mt[i] ^ ((mt[i - 1] ^ (mt[i - 1] >> 30)) * 1566083941u)) - (unsigned)i;
    ++i;
    if (i >= 624) { mt[0] = mt[623]; i = 1; }
  }
  mt[0] = 0x80000000u;
  int mti = 624;
  unsigned sel[NPTS / 32];                             // selection-set branch
  for (int t = 0; t < NPTS / 32; ++t) sel[t] = 0u;
  for (int t = 0; t < KC; ++t) {
    unsigned r;
    for (;;) {
      r = mt_next(mt, mti) >> 17;                      // getrandbits(15)
      if (r >= (unsigned)NPTS) continue;               // _randbelow rejection
      if ((sel[r >> 5] >> (r & 31u)) & 1u) continue;   // "while j in selected"
      break;
    }
    sel[r >> 5] |= 1u << (r & 31u);
    inds[t] = (int)r;
  }
}

// v0[b][k][:] = x[b][inds[k]][:]
__global__ void k_gather(const float* __restrict__ x, const int* __restrict__ inds,
                         float* __restrict__ v) {
  int k = blockIdx.x, b = blockIdx.y, d = threadIdx.x;
  v[((size_t)b * KC + k) * DIM + d] = x[((size_t)b * NPTS + inds[k]) * DIM + d];
}

// out[p] = sum_d in[p][d]^2
__global__ void k_rownorm(const float* __restrict__ x, float* __restrict__ out, int nrows) {
  int p = blockIdx.x * blockDim.x + threadIdx.x;
  if (p >= nrows) return;
  const float4* row = (const float4*)(x + (size_t)p * DIM);
  float s = 0.f;
  for (int t = 0; t < DIM / 4; ++t) {
    float4 q = row[t];
    s += q.x * q.x + q.y * q.y + q.z * q.z + q.w * q.w;
  }
  out[p] = s;
}

// T[b][d] = sum_n x[b][n][d]
__global__ void k_colsum(const float* __restrict__ x, float* __restrict__ T) {
  int b = blockIdx.x, d = threadIdx.x;
  const float* xb = x + (size_t)b * NPTS * DIM;
  float s = 0.f;
  for (int n = 0; n < NPTS; ++n) s += xb[(size_t)n * DIM + d];
  T[b * DIM + d] = s;
}

// ---------------------------------------------------------------------------
// Async-to-LDS tile stage: rows x 256 f32, b128 chunks, ASYNCcnt tracked
// ---------------------------------------------------------------------------
__device__ __forceinline__ void async_stage(const float* __restrict__ g,
                                            float (*s)[PITCH], int rows, int tid) {
  for (int idx = tid; idx < rows * (DIM / 4); idx += TPBA) {
    int r = idx >> 6, c4 = idx & 63;
    unsigned           lds = (unsigned)(size_t)&s[r][c4 * 4];
    unsigned long long ga  = (unsigned long long)(size_t)(g + (size_t)r * DIM + c4 * 4);
    asm volatile("global_load_async_to_lds_b128 %0, %1, off"
                 :: "v"(lds), "v"(ga) : "memory");
  }
}

// ---------------------------------------------------------------------------
// Assignment: argmin_k relu(x2 - 2 x.v + v2) via V_WMMA_F32_16X16X4_F32.
// 2 cluster-tiles per wave (one A fragment feeds two WMMAs), double-buffered
// async v-tile staging overlapped with compute.
// ---------------------------------------------------------------------------
__global__ __launch_bounds__(TPBA) void k_assign(const float* __restrict__ x,
                                                 const float* __restrict__ v,
                                                 const float* __restrict__ x2,
                                                 const float* __restrict__ v2,
                                                 int* __restrict__ asg) {
  __shared__ float sx[ROWS][PITCH];     // 64 x 258 f32  (~66 KB)
  __shared__ float sv[2][32][PITCH];    // double-buffered 32-cluster tiles (~66 KB)
  __shared__ float sx2[ROWS];

  const int b    = blockIdx.y;
  const int row0 = blockIdx.x * ROWS;
  const int tid  = threadIdx.x;
  const int lane = tid & 31;
  const int wave = tid >> 5;
  const int col  = lane & 15;   // A: M index / B: N index
  const int hh   = lane >> 4;   // K offset selector {0,2} per ISA A/B layouts
  const int wr   = wave * 16;   // wave's 16-row slice of the x tile

  // Stage x tile (async) + x2 slice
  async_stage(x + ((size_t)b * NPTS + row0) * DIM, sx, ROWS, tid);
  if (tid < ROWS) sx2[tid] = x2[(size_t)b * NPTS + row0 + tid];

  // Prologue: stage first 32-cluster v tile into buffer 0
  const float* vbase = v + (size_t)b * KC * DIM;
  async_stage(vbase, sv[0], 32, tid);

  float bestv[8];
  int   besti[8];
#pragma unroll
  for (int r = 0; r < 8; ++r) { bestv[r] = 3.4e38f; besti[r] = 0x7fffffff; }

  const float* v2b = v2 + (size_t)b * KC;

  for (int t = 0; t < KC / 32; ++t) {             // 16 iterations x 32 clusters
    asm volatile("s_wait_asynccnt 0" ::: "memory");  // my async writes landed
    __syncthreads();                                  // everyone's landed; prev compute done
    if (t + 1 < KC / 32)                              // overlap next tile with compute
      async_stage(vbase + (size_t)(t + 1) * 32 * DIM, sv[(t + 1) & 1], 32, tid);

    const float(*svc)[PITCH] = sv[t & 1];

    v8f c0 = {0.f, 0.f, 0.f, 0.f, 0.f, 0.f, 0.f, 0.f};
    v8f c1 = c0;
#pragma unroll 8
    for (int d = 0; d < DIM; d += 4) {
      v2f a  = *(const v2f*)&sx[wr + col][d + 2 * hh];   // shared A fragment
      v2f b0 = *(const v2f*)&svc[col][d + 2 * hh];
      v2f b1 = *(const v2f*)&svc[16 + col][d + 2 * hh];
      c0 = __builtin_amdgcn_wmma_f32_16x16x4_f32(false, a, false, b0, (short)0, c0, false, false);
      c1 = __builtin_amdgcn_wmma_f32_16x16x4_f32(false, a, false, b1, (short)0, c1, false, false);
    }

    const float v2a = v2b[t * 32 + col];
    const float v2c = v2b[t * 32 + 16 + col];
    const int   cl0 = t * 32 + col;
    const int   cl1 = t * 32 + 16 + col;
#pragma unroll
    for (int r = 0; r < 8; ++r) {
      float xx = sx2[wr + r + 8 * hh];
      float d0 = fmaxf(0.f, xx - 2.f * c0[r] + v2a);
      if (d0 < bestv[r]) { bestv[r] = d0; besti[r] = cl0; }   // lower index first
      float d1 = fmaxf(0.f, xx - 2.f * c1[r] + v2c);
      if (d1 < bestv[r]) { bestv[r] = d1; besti[r] = cl1; }
    }
  }

  // Reduce the 16 column slots per row (masks <=8 stay inside each 16-lane half)
#pragma unroll
  for (int r = 0; r < 8; ++r) {
    float bv = bestv[r];
    int   bi = besti[r];
#pragma unroll
    for (int m = 1; m <= 8; m <<= 1) {
      float ov = __shfl_xor(bv, m, 32);
      int   oi = __shfl_xor(bi, m, 32);
      if (ov < bv || (ov == bv && oi < bi)) { bv = ov; bi = oi; }
    }
    if (col == 0)
      asg[(size_t)b * NPTS + row0 + wr + r + 8 * hh] = bi;
  }
}

// Scatter-add x into centroid accumulators (4 points per block)
__global__ void k_accum(const float* __restrict__ x, const int* __restrict__ asg,
                        float* __restrict__ vsum, float* __restrict__ counts) {
  int base = blockIdx.x * 4;
  int d = threadIdx.x;
#pragma unroll
  for (int i = 0; i < 4; ++i) {
    int p = base + i;
    int b = p >> 14;
    int a = asg[p];
    atomicAdd(&vsum[((size_t)b * KC + a) * DIM + d], x[(size_t)p * DIM + d]);
    if (d == 0) atomicAdd(&counts[b * KC + a], 1.0f);
  }
}

// v[k][d] = (S_k[d] + eps*T[d]) / (count_k + eps)
__global__ void k_vupdate(const float* __restrict__ vsum, const float* __restrict__ counts,
                          const float* __restrict__ T, float* __restrict__ v) {
  int bk = blockIdx.x, d = threadIdx.x, b = bk >> 9;
  v[(size_t)bk * DIM + d] =
      (vsum[(size_t)bk * DIM + d] + EPSF * T[b * DIM + d]) / (counts[bk] + EPSF);
}

// Expand final assignments into one-hot u
__global__ void k_write_u(const int* __restrict__ asg, float* __restrict__ out) {
  int p = blockIdx.x;
  int t = threadIdx.x;          // 128 threads, float4 each -> 512 cols
  int a = asg[p];
  int c0 = t * 4;
  float4 r;
  r.x = (a == c0 + 0) ? 1.f : 0.f;
  r.y = (a == c0 + 1) ? 1.f : 0.f;
  r.z = (a == c0 + 2) ? 1.f : 0.f;
  r.w = (a == c0 + 3) ? 1.f : 0.f;
  ((float4*)(out + (size_t)p * KC))[t] = r;
}

__global__ void k_copyv(const float* __restrict__ v, float* __restrict__ out) {
  int bk = blockIdx.x, d = threadIdx.x;
  out[(size_t)bk * DIM + d] = v[(size_t)bk * DIM + d];
}

// ---------------------------------------------------------------------------
extern "C" void kernel_launch(void* const* d_in, const int* in_sizes, int n_in,
                              void* d_out, int out_size, void* d_ws, size_t ws_size,
                              hipStream_t stream) {
  (void)in_sizes; (void)n_in; (void)out_size; (void)ws_size;
  const float* x = (const float*)d_in[0];
  float* out = (float*)d_out;
  char* ws = (char*)d_ws;

  // Workspace carving (all offsets 16B aligned)
  int*   inds   = (int*)(ws);                                       // 512 ints
  int*   asg    = (int*)(ws + 2048);                                // 65536 ints
  float* x2     = (float*)(ws + 2048 + 262144);                     // 65536 f
  float* T      = (float*)(ws + 2048 + 262144 + 262144);            // 1024 f
  float* v      = (float*)(ws + 2048 + 262144 + 262144 + 4096);     // 524288 f
  float* v2     = (float*)((char*)v + 2097152);                     // 2048 f
  float* vsum   = (float*)((char*)v2 + 8192);                       // 524288 f
  float* counts = (float*)((char*)vsum + 2097152);                  // 2048 f (contiguous w/ vsum)

  k_init_inds<<<1, 1, 0, stream>>>(inds);
  k_gather<<<dim3(KC, BATCH), DIM, 0, stream>>>(x, inds, v);
  k_rownorm<<<(BATCH * NPTS) / 256, 256, 0, stream>>>(x, x2, BATCH * NPTS);
  k_colsum<<<BATCH, DIM, 0, stream>>>(x, T);

  for (int it = 0; it < NITER; ++it) {
    k_rownorm<<<(BATCH * KC) / 256, 256, 0, stream>>>(v, v2, BATCH * KC);
    k_assign<<<dim3(NPTS / ROWS, BATCH), TPBA, 0, stream>>>(x, v, x2, v2, asg);
    hipMemsetAsync(vsum, 0, (size_t)(BATCH * KC * DIM + BATCH * KC) * sizeof(float), stream);
    k_accum<<<(BATCH * NPTS) / 4, DIM, 0, stream>>>(x, asg, vsum, counts);
    k_vupdate<<<BATCH * KC, DIM, 0, stream>>>(vsum, counts, T, v);
  }

  k_write_u<<<BATCH * NPTS, KC / 4, 0, stream>>>(asg, out);
  k_copyv<<<BATCH * KC, DIM, 0, stream>>>(v, out + (size_t)BATCH * NPTS * KC);
}